// HSM_90589450207921
// MI455X (gfx1250) — compile-verified
//
#include <hip/hip_runtime.h>

// Problem constants (match reference)
#define B_ROWS 4096
#define N_COLS 32768
#define L_LVL  12
#define K_BR   64

typedef __attribute__((ext_vector_type(2))) float v2f;
typedef __attribute__((ext_vector_type(8))) float v8f;

// ---------------------------------------------------------------------------
// Kernel 1: one block per row b, 12 waves (384 threads), wave l handles level l.
// Each lane gathers 2 of the 64 columns for (b, l); wave32 shuffle reductions
// compute LSE; contribution = LSE - x[b, p_y[l]].  Per-row total -> partial[b].
// ---------------------------------------------------------------------------
__global__ __launch_bounds__(L_LVL * 32) void hsm_rows(
    const float* __restrict__ x,
    const int*   __restrict__ brother,   // [L_LVL * K_BR]
    const int*   __restrict__ p_y,       // [L_LVL]
    float*       __restrict__ partial)   // [B_ROWS]
{
    __shared__ float lds_lvl[L_LVL];

    const int b    = blockIdx.x;
    const int lane = threadIdx.x & 31;
    const int l    = threadIdx.x >> 5;          // wave index == level

    // Prefetch the tiny hot index tables (3 KB) into cache: 24 lanes each
    // prefetch one 128B line (emits global_prefetch_b8, gfx1250 path).
    if (threadIdx.x < (L_LVL * K_BR) / 32)
        __builtin_prefetch(&brother[threadIdx.x * 32], 0, 0);
    if (threadIdx.x == 0)
        __builtin_prefetch(p_y, 0, 0);

    const float* __restrict__ xrow = x + (size_t)b * N_COLS;

    // Gather: 2 columns per lane (K=64 over wave32).
    const int i0 = brother[l * K_BR + lane];
    const int i1 = brother[l * K_BR + 32 + lane];
    const float v0 = xrow[i0];
    const float v1 = xrow[i1];
    const float py = xrow[p_y[l]];              // uniform per wave

    // Wave max reduction (log-sum-exp stabilization).
    float m = fmaxf(v0, v1);
    #pragma unroll
    for (int off = 16; off > 0; off >>= 1)
        m = fmaxf(m, __shfl_xor(m, off, 32));

    // Wave sum of exp.
    float s = expf(v0 - m) + expf(v1 - m);
    #pragma unroll
    for (int off = 16; off > 0; off >>= 1)
        s += __shfl_xor(s, off, 32);

    if (lane == 0)
        lds_lvl[l] = (m + logf(s)) - py;        // -logp[b,l,label] = lse - x[b,p_y[l]]

    __syncthreads();

    if (threadIdx.x == 0) {
        float t = 0.0f;
        #pragma unroll
        for (int i = 0; i < L_LVL; ++i) t += lds_lvl[i];
        partial[b] = t;
    }
}

// ---------------------------------------------------------------------------
// Kernel 2: deterministic reduction of 4096 partials using WMMA.
// A = 16x4 all-ones, so D = A*B + C accumulates the column sums of each
// 64-element chunk (every chunk element appears exactly once in B).
// 64 chained v_wmma_f32_16x16x4_f32 ops; all 16 rows of D are identical, so
// sum row 0 (c[0] on lanes 0..15) and scale by 1/B.
// Single wave of 32 threads, no divergence before WMMA (EXEC all ones).
// ---------------------------------------------------------------------------
__global__ __launch_bounds__(32) void hsm_reduce(
    const float* __restrict__ partial,
    float*       __restrict__ out)
{
    const int lane = threadIdx.x;

    v2f a; a.x = 1.0f; a.y = 1.0f;              // A matrix = all ones
    v8f c = {};

    for (int i = 0; i < B_ROWS; i += 64) {
        v2f bb;
        bb.x = partial[i + lane];
        bb.y = partial[i + 32 + lane];
        c = __builtin_amdgcn_wmma_f32_16x16x4_f32(
                /*neg_a=*/false, a, /*neg_b=*/false, bb,
                /*c_mod=*/(short)0, c, /*reuse_a=*/false, /*reuse_b=*/false);
    }

    // Sum D[0, 0..15]: lanes 0..15 of c[0] (xor tree over low 4 lane bits).
    float t = c[0];
    t += __shfl_xor(t, 1, 32);
    t += __shfl_xor(t, 2, 32);
    t += __shfl_xor(t, 4, 32);
    t += __shfl_xor(t, 8, 32);

    if (lane == 0)
        out[0] = t * (1.0f / (float)B_ROWS);
}

// ---------------------------------------------------------------------------
extern "C" void kernel_launch(void* const* d_in, const int* in_sizes, int n_in,
                              void* d_out, int out_size, void* d_ws, size_t ws_size,
                              hipStream_t stream)
{
    const float* x       = (const float*)d_in[0];   // (B, N) f32
    const int*   brother = (const int*)  d_in[1];   // (L, K) i32
    const int*   p_y     = (const int*)  d_in[2];   // (L,)   i32
    // d_in[3] = y : unused by the reference output.

    float* out     = (float*)d_out;                 // scalar f32
    float* partial = (float*)d_ws;                  // B_ROWS floats of scratch

    hsm_rows<<<B_ROWS, L_LVL * 32, 0, stream>>>(x, brother, p_y, partial);
    hsm_reduce<<<1, 32, 0, stream>>>(partial, out);
}